// SSIM3D_26096221290788
// MI455X (gfx1250) — compile-verified
//
#include <hip/hip_runtime.h>

typedef float v2f __attribute__((ext_vector_type(2)));
typedef float v8f __attribute__((ext_vector_type(8)));

namespace {
constexpr int kT = 16, kH = 256, kW = 256;
constexpr int kBC = 6;                 // B*C
constexpr int TZ = 22;                 // full T + 6 halo (covers all t)
constexpr int KPAD = 24;               // K padded to 6 chunks of 4 (rows 22,23 zero)
constexpr int HT = 16, WT = 32;        // output tile (h x w)
constexpr int HZ = HT + 6;             // 22 rows with h-halo
constexpr int WLD = 40;                // aligned load span: gw in [w0-4, w0+36)
constexpr int ROWS = TZ * HZ;          // 484 (t,h) rows in tempW
constexpr int TW_SZ = ROWS * WT;       // 15488 floats
constexpr int COLS = HT * WT;          // 512 columns for T-conv
constexpr int TH_SZ = TZ * COLS;       // 11264 floats written per product
constexpr int TH_PAD_SZ = KPAD * COLS; // 12288 floats allocated (incl. zero pad rows)
constexpr int NTHREADS = 512;          // 16 waves (wave32)
constexpr int NWAVES = NTHREADS / 32;
constexpr int GRP_PER_WAVE = COLS / (NWAVES * 16); // 2
constexpr int NBLK = kBC * (kH / HT) * (kW / WT);  // 768
constexpr float C1c = 1.0e-4f, C2c = 9.0e-4f;
constexpr float kInvN = 1.0f / 6291456.0f;
}

__global__ __launch_bounds__(NTHREADS) void ssim3d_fused(
    const float* __restrict__ X, const float* __restrict__ Y,
    float* __restrict__ partial)
{
  extern __shared__ float smem[];
  float* tW  = smem;                       // [ROWS][WT]        W-conv result
  float* tH  = smem + TW_SZ;               // [KPAD][COLS]      H-conv result (+2 zero rows)
  float* red = smem + TW_SZ + TH_PAD_SZ;   // [NTHREADS]        reduction

  const int tid  = threadIdx.x;
  const int lane = tid & 31;
  const int wv   = tid >> 5;

  // Gaussian taps, sigma = 1.5, normalized (matches reference).
  // g[] is only ever indexed with compile-time constants (stages 1-2).
  float g[7];
  float gnorm;
  {
    float s = 0.f;
#pragma unroll
    for (int i = 0; i < 7; ++i) {
      float d = (float)(i - 3);
      g[i] = __expf(-(d * d) * (1.0f / 4.5f));
      s += g[i];
    }
    gnorm = 1.0f / s;
#pragma unroll
    for (int i = 0; i < 7; ++i) g[i] *= gnorm;
  }

  // Zero the K-padding rows of tH once; stage 2 never touches rows 22..23,
  // so they remain zero for all five products.
  for (int o = tid; o < 2 * COLS; o += NTHREADS) tH[TH_SZ + o] = 0.0f;

  const int blk = blockIdx.x;
  const int bc  = blk >> 7;          // 128 tiles per (b,c)
  const int rem = blk & 127;
  const int h0  = (rem >> 3) * HT;
  const int w0  = (rem & 7) * WT;

  const float* Xb = X + (size_t)bc * kT * kH * kW;
  const float* Yb = Y + (size_t)bc * kT * kH * kW;

  // Constant banded A fragments for T-conv: A(m,k) = g[k-m] when 0<=k-m<=6.
  // Band value computed analytically (gnorm * exp(-(d-3)^2/4.5)) so a
  // per-lane-varying index never gathers from the register array g[]
  // (avoids compare/select cascades and makes remat cheap).
  const int m   = lane & 15;
  const int dlt = (lane < 16) ? 0 : 2;
  v2f afrag[6];
#pragma unroll
  for (int c = 0; c < 6; ++c) {
    const int k0 = 4 * c + dlt;
    const int d0 = k0 - m, d1 = k0 + 1 - m;
    const float t0 = (float)(d0 - 3), t1 = (float)(d1 - 3);
    const float v0 = gnorm * __expf(-(t0 * t0) * (1.0f / 4.5f));
    const float v1 = gnorm * __expf(-(t1 * t1) * (1.0f / 4.5f));
    afrag[c].x = ((unsigned)d0 <= 6u) ? v0 : 0.0f;
    afrag[c].y = ((unsigned)d1 <= 6u) ? v1 : 0.0f;
  }

  v8f acc[5][GRP_PER_WAVE];

  __syncthreads();  // tH pad rows visible to all waves

#pragma unroll
  for (int p = 0; p < 5; ++p) {
    // ---- stage 1: pointwise product + W-direction 7-tap conv, global -> tempW ----
    for (int r = tid; r < ROWS; r += NTHREADS) {
      const int tz = r / HZ;
      const int hz = r - tz * HZ;
      const int gt = tz - 3;
      const int gh = h0 + hz - 3;
      const bool rowok = ((unsigned)gt < (unsigned)kT) && ((unsigned)gh < (unsigned)kH);
      const float* xrow = Xb + ((size_t)gt * kH + gh) * kW;
      const float* yrow = Yb + ((size_t)gt * kH + gh) * kW;

      // xa[i]/ya[i] cover gw = w0 - 4 + i, i in [0, 40): one element wider
      // than the conv needs so the fast path is 16B-aligned ((w0-4)*4 % 16 == 0).
      float xa[WLD], ya[WLD];
      if (rowok && w0 >= 4 && (w0 + WLD - 4) <= kW) {
        // Interior fast path: 10+10 aligned float4 loads -> global_load_b128.
        const float4* xv = (const float4*)(xrow + (w0 - 4));
        const float4* yv = (const float4*)(yrow + (w0 - 4));
#pragma unroll
        for (int q = 0; q < WLD / 4; ++q) {
          float4 a = xv[q];
          float4 b = yv[q];
          xa[4 * q + 0] = a.x; xa[4 * q + 1] = a.y; xa[4 * q + 2] = a.z; xa[4 * q + 3] = a.w;
          ya[4 * q + 0] = b.x; ya[4 * q + 1] = b.y; ya[4 * q + 2] = b.z; ya[4 * q + 3] = b.w;
        }
      } else {
#pragma unroll
        for (int i = 0; i < WLD; ++i) {
          int gw = w0 - 4 + i;
          bool ok = rowok && ((unsigned)gw < (unsigned)kW);
          xa[i] = ok ? xrow[gw] : 0.0f;
          ya[i] = ok ? yrow[gw] : 0.0f;
        }
      }

      float pa[WLD];
#pragma unroll
      for (int i = 0; i < WLD; ++i) {
        pa[i] = (p == 0) ? xa[i]
              : (p == 1) ? ya[i]
              : (p == 2) ? xa[i] * xa[i]
              : (p == 3) ? ya[i] * ya[i]
                         : xa[i] * ya[i];
      }
#pragma unroll
      for (int w = 0; w < WT; ++w) {
        float a = 0.0f;
#pragma unroll
        for (int j = 0; j < 7; ++j) a = fmaf(g[j], pa[w + j + 1], a);  // +1: xa base is w0-4
        tW[r * WT + w] = a;
      }
    }
    __syncthreads();

    // ---- stage 2: H-direction 7-tap conv, tempW -> tempH ----
    for (int o = tid; o < TH_SZ; o += NTHREADS) {
      const int w  = o & (WT - 1);
      const int hy = (o / WT) & (HT - 1);
      const int tz = o / COLS;
      float a = 0.0f;
#pragma unroll
      for (int j = 0; j < 7; ++j)
        a = fmaf(g[j], tW[(tz * HZ + hy + j) * WT + w], a);
      tH[o] = a;  // o == tz*COLS + (hy*WT + w)
    }
    __syncthreads();

    // ---- stage 3: T-direction conv as banded GEMM via V_WMMA_F32_16X16X4_F32 ----
    // D(t,n) = sum_k Band(t,k) * tH(k, col_n); K = 24 (rows 22,23 zero pad).
    // Preload all six B fragments, then issue the WMMA chain back-to-back.
#pragma unroll
    for (int grp = 0; grp < GRP_PER_WAVE; ++grp) {
      const int col = (wv * GRP_PER_WAVE + grp) * 16 + m;   // 0..COLS-1
      v2f barr[6];
#pragma unroll
      for (int c = 0; c < 6; ++c) {
        const int k0 = 4 * c + dlt;
        barr[c].x = tH[(k0    ) * COLS + col];
        barr[c].y = tH[(k0 + 1) * COLS + col];
      }
      v8f d = {0.f, 0.f, 0.f, 0.f, 0.f, 0.f, 0.f, 0.f};
#pragma unroll
      for (int c = 0; c < 6; ++c) {
        d = __builtin_amdgcn_wmma_f32_16x16x4_f32(
            false, afrag[c], false, barr[c], (short)0, d, false, false);
      }
      acc[p][grp] = d;
    }
    __syncthreads();  // tempW/tempH reused by next product
  }

  // ---- stage 4: SSIM map from WMMA accumulators + block reduction ----
  float lsum = 0.0f;
#pragma unroll
  for (int grp = 0; grp < GRP_PER_WAVE; ++grp) {
#pragma unroll
    for (int r = 0; r < 8; ++r) {
      float mu1 = acc[0][grp][r];
      float mu2 = acc[1][grp][r];
      float exx = acc[2][grp][r];
      float eyy = acc[3][grp][r];
      float exy = acc[4][grp][r];
      float mu1s = mu1 * mu1, mu2s = mu2 * mu2, m12 = mu1 * mu2;
      float s1 = exx - mu1s, s2 = eyy - mu2s, s12 = exy - m12;
      float num = (2.0f * m12 + C1c) * (2.0f * s12 + C2c);
      float den = (mu1s + mu2s + C1c) * (s1 + s2 + C2c);
      lsum += 1.0f - num / den;
    }
  }
  red[tid] = lsum;
  __syncthreads();
#pragma unroll
  for (int s = NTHREADS / 2; s > 0; s >>= 1) {
    if (tid < s) red[tid] += red[tid + s];
    __syncthreads();
  }
  if (tid == 0) partial[blk] = red[0];
}

__global__ __launch_bounds__(256) void ssim3d_reduce(
    const float* __restrict__ partial, float* __restrict__ out, int n, float invN)
{
  __shared__ float red[256];
  float s = 0.0f;
  for (int i = threadIdx.x; i < n; i += 256) s += partial[i];
  red[threadIdx.x] = s;
  __syncthreads();
  for (int k = 128; k > 0; k >>= 1) {
    if (threadIdx.x < k) red[threadIdx.x] += red[threadIdx.x + k];
    __syncthreads();
  }
  if (threadIdx.x == 0) out[0] = red[0] * invN;
}

extern "C" void kernel_launch(void* const* d_in, const int* in_sizes, int n_in,
                              void* d_out, int out_size, void* d_ws, size_t ws_size,
                              hipStream_t stream) {
  const float* X = (const float*)d_in[0];   // input  (2,3,16,256,256) f32
  const float* Y = (const float*)d_in[1];   // target (2,3,16,256,256) f32
  float* out = (float*)d_out;               // scalar f32
  float* partial = (float*)d_ws;            // NBLK floats of scratch

  size_t shmem = (size_t)(TW_SZ + TH_PAD_SZ + NTHREADS) * sizeof(float); // ~111 KB of 320 KB LDS
  ssim3d_fused<<<NBLK, NTHREADS, shmem, stream>>>(X, Y, partial);
  ssim3d_reduce<<<1, 256, 0, stream>>>(partial, out, NBLK, kInvN);

  (void)in_sizes; (void)n_in; (void)out_size; (void)ws_size;
}